// SecondEncoder_1941325218151
// MI455X (gfx1250) — compile-verified
//
#include <hip/hip_runtime.h>

typedef float v2f __attribute__((ext_vector_type(2)));
typedef float v8f __attribute__((ext_vector_type(8)));

#define DFEAT 128

// ---------------- utility kernels ----------------

__global__ void k_fill_zero(float* __restrict__ p, int n) {
    int i = blockIdx.x * blockDim.x + threadIdx.x;
    if (i < n) p[i] = 0.0f;
}

__global__ void k_deg_count(const long long* __restrict__ col, float* __restrict__ deg, int e) {
    int i = blockIdx.x * blockDim.x + threadIdx.x;
    if (i < e) atomicAdd(&deg[(int)col[i]], 1.0f);
}

// deg -> deg^{-1/2} including self loop (+1)
__global__ void k_dis(float* __restrict__ deg, int n) {
    int i = blockIdx.x * blockDim.x + threadIdx.x;
    if (i < n) deg[i] = rsqrtf(deg[i] + 1.0f);
}

__global__ void k_norm(const long long* __restrict__ row, const long long* __restrict__ col,
                       const float* __restrict__ dis, float* __restrict__ norm, int e) {
    int i = blockIdx.x * blockDim.x + threadIdx.x;
    if (i < e) norm[i] = dis[(int)row[i]] * dis[(int)col[i]];
}

// ---------------- fp32 WMMA GEMM: H[N,128] = X[N,128] @ W[128,128] ----------------
// One wave32 computes one 16x16 output tile via V_WMMA_F32_16X16X4_F32, K-loop of 32 steps.
__global__ void k_gemm_wmma(const float* __restrict__ X, const float* __restrict__ W,
                            float* __restrict__ H, int nrows) {
    const int lane = threadIdx.x & 31;
    const int wave = threadIdx.x >> 5;
    const int gw   = blockIdx.x * 8 + wave;          // 8 waves / block
    const int tileM = gw >> 3;                        // D/16 = 8 tiles in N-dim
    const int tileN = gw & 7;
    if (tileM * 16 >= nrows) return;

    const int half = lane >> 4;                       // 0: lanes 0-15, 1: lanes 16-31
    const int l15  = lane & 15;
    const int arow = tileM * 16 + l15;                // A: M striped over lanes 0-15 / 16-31
    const int bcol = tileN * 16 + l15;                // B: N striped over lanes

    v8f acc = {};
#pragma unroll 4
    for (int kk = 0; kk < DFEAT; kk += 4) {
        const int ka = kk + half * 2;                 // lanes 16-31 hold K=2,3
        // A 16x4: VGPR0=K ka, VGPR1=K ka+1 (contiguous in row-major X)
        float2 at = *(const float2*)(X + (size_t)arow * DFEAT + ka);
        v2f a; a.x = at.x; a.y = at.y;
        // B 4x16: VGPR0 = W[ka][col], VGPR1 = W[ka+1][col]
        v2f b;
        b.x = W[(size_t)ka * DFEAT + bcol];
        b.y = W[(size_t)(ka + 1) * DFEAT + bcol];
        acc = __builtin_amdgcn_wmma_f32_16x16x4_f32(false, a, false, b,
                                                    (short)0, acc, false, false);
    }

    // D layout: VGPR v -> row tileM*16 + half*8 + v, col = bcol
    const int rbase = tileM * 16 + half * 8;
#pragma unroll
    for (int v = 0; v < 8; ++v)
        H[(size_t)(rbase + v) * DFEAT + bcol] = acc[v];
}

// ---------------- self-loop init: A = H * dis[i]^2 ----------------
__global__ void k_selfloop_init(const float* __restrict__ H, const float* __restrict__ dis,
                                float* __restrict__ A, int total) {
    int i = blockIdx.x * blockDim.x + threadIdx.x;
    if (i < total) {
        float s = dis[i >> 7];                        // i / 128
        A[i] = H[i] * (s * s);
    }
}

// ---------------- edge scatter: one wave per edge, float4 per lane ----------------
__global__ void k_scatter(const float* __restrict__ H, const long long* __restrict__ row,
                          const long long* __restrict__ col, const float* __restrict__ norm,
                          float* __restrict__ A, int e) {
    const int lane = threadIdx.x & 31;
    const int widx = (blockIdx.x * blockDim.x + threadIdx.x) >> 5;
    if (widx >= e) return;
    const int r = (int)row[widx];
    const int c = (int)col[widx];
    const float nv = norm[widx];
    float4 m = *((const float4*)(H + (size_t)r * DFEAT) + lane);   // 512B/edge gather
    float* dst = A + (size_t)c * DFEAT + lane * 4;
    atomicAdd(dst + 0, m.x * nv);
    atomicAdd(dst + 1, m.y * nv);
    atomicAdd(dst + 2, m.z * nv);
    atomicAdd(dst + 3, m.w * nv);
}

// ---------------- bias (layer 1 output) ----------------
__global__ void k_bias(const float* __restrict__ A, const float* __restrict__ b,
                       float* __restrict__ out, int total) {
    int i = blockIdx.x * blockDim.x + threadIdx.x;
    if (i < total) out[i] = A[i] + b[i & (DFEAT - 1)];
}

// ---------------- bias + leaky_relu(0.1) (final output) ----------------
__global__ void k_bias_lrelu(const float* __restrict__ A, const float* __restrict__ b,
                             float* __restrict__ out, int total) {
    int i = blockIdx.x * blockDim.x + threadIdx.x;
    if (i < total) {
        float v = A[i] + b[i & (DFEAT - 1)];
        out[i] = v > 0.0f ? v : 0.1f * v;
    }
}

extern "C" void kernel_launch(void* const* d_in, const int* in_sizes, int n_in,
                              void* d_out, int out_size, void* d_ws, size_t ws_size,
                              hipStream_t stream) {
    const float*     x  = (const float*)d_in[0];
    const long long* ei = (const long long*)d_in[1];      // int64 [2, E]
    const float*     W1 = (const float*)d_in[2];
    const float*     b1 = (const float*)d_in[3];
    const float*     W2 = (const float*)d_in[4];
    const float*     b2 = (const float*)d_in[5];
    float* out = (float*)d_out;

    const int N = in_sizes[0] / DFEAT;
    const int E = in_sizes[1] / 2;
    const int ND = N * DFEAT;

    const long long* erow = ei;          // sources
    const long long* ecol = ei + E;      // targets

    // workspace layout (floats): dis[N] | norm[E] | G[N*D] | A[N*D]
    float* ws  = (float*)d_ws;
    float* dis = ws;
    float* nrm = ws + N;
    float* G   = nrm + E;
    float* A   = G + ND;

    const int B = 256;
    const int gN  = (N  + B - 1) / B;
    const int gE  = (E  + B - 1) / B;
    const int gND = (ND + B - 1) / B;
    const int gGemm = (N + 15) / 16;          // 8 waves/block * 8 N-tiles == tilesM blocks
    const int gScat = (E + 7) / 8;            // 8 edges (waves) per 256-thread block

    // degree normalization (shared by both layers)
    k_fill_zero<<<gN, B, 0, stream>>>(dis, N);
    k_deg_count<<<gE, B, 0, stream>>>(ecol, dis, E);
    k_dis<<<gN, B, 0, stream>>>(dis, N);
    k_norm<<<gE, B, 0, stream>>>(erow, ecol, dis, nrm, E);

    // ---- layer 1 ----
    k_gemm_wmma<<<gGemm, B, 0, stream>>>(x, W1, G, N);
    k_selfloop_init<<<gND, B, 0, stream>>>(G, dis, A, ND);
    k_scatter<<<gScat, B, 0, stream>>>(G, erow, ecol, nrm, A, E);
    k_bias<<<gND, B, 0, stream>>>(A, b1, G, ND);          // G = layer-1 activations

    // ---- layer 2 ----
    k_gemm_wmma<<<gGemm, B, 0, stream>>>(G, W2, A, N);    // A = G @ W2
    k_selfloop_init<<<gND, B, 0, stream>>>(A, dis, G, ND);// G = self-loop term
    k_scatter<<<gScat, B, 0, stream>>>(A, erow, ecol, nrm, G, E);
    k_bias_lrelu<<<gND, B, 0, stream>>>(G, b2, out, ND);
}